// DiversityLoss_56530359550796
// MI455X (gfx1250) — compile-verified
//
#include <hip/hip_runtime.h>

// DiversityLoss on MI455X (gfx1250, wave32).
// out = (sum(G) - B*trace(G)) / (D * B * (B-1)),  G = x x^T, x: [64, 196608] fp32.
// Single pass over HBM (~50 MB); Gram partials computed on the matrix pipe via
// V_WMMA_F32_16X16X4_F32. trace(G) = sum(x^2) is accumulated for free during the
// global->LDS staging loop (data already in registers), so no accumulator
// extraction / scratch spills are needed.

#define B_IMG   64
#define D_ELEM  196608            // 3*256*256
#define CK      128               // K-chunk of columns per workgroup
#define NWG     (D_ELEM / CK)     // 1536 workgroups
#define STRIDE  (CK + 4)          // 132 floats/row; stride % 64 == 4 -> conflict-free b64 reads
#define THREADS 256               // 8 waves of 32

typedef float v2f __attribute__((ext_vector_type(2)));
typedef float v4f __attribute__((ext_vector_type(4)));
typedef float v8f __attribute__((ext_vector_type(8)));

__global__ __launch_bounds__(THREADS)
void gram_partial_kernel(const float* __restrict__ x, float* __restrict__ ws) {
  __shared__ float lds[B_IMG * STRIDE];   // 64*132*4 = 33792 B
  __shared__ float redA[THREADS];
  __shared__ float redB[THREADS];

  const int tid  = threadIdx.x;
  const int wg   = blockIdx.x;
  const int lane = tid & 31;
  const int wave = tid >> 5;
  const long long k0g = (long long)wg * CK;

  // ---- Stage chunk: 64 rows x 128 floats, global -> LDS via b128.
  //      Accumulate trace partial (sum of squares) on the fly.
  float tr = 0.f;
  #pragma unroll
  for (int i = 0; i < (B_IMG * CK) / (THREADS * 4); ++i) {  // 8 iterations
    int j   = i * THREADS + tid;        // float4 index; 32 float4 per row
    int row = j >> 5;
    int col = (j & 31) * 4;
    v4f v = *(const v4f*)(x + (long long)row * D_ELEM + k0g + col);
    tr = fmaf(v.x, v.x, tr);
    tr = fmaf(v.y, v.y, tr);
    tr = fmaf(v.z, v.z, tr);
    tr = fmaf(v.w, v.w, tr);
    *(v4f*)(&lds[row * STRIDE + col]) = v;
  }
  __syncthreads();

  // ---- WMMA: each wave accumulates two 16x16 tiles of G = X * X^T ----
  // Tiles 2w and 2w+1 share the same 16 A-rows.
  const int t0   = 2 * wave;
  const int m0   = (t0 >> 2) * 16;        // A row block
  const int n00  = (t0 & 3) * 16;         // B row block, tile 0
  const int n01  = ((t0 + 1) & 3) * 16;   // B row block, tile 1
  const int l15  = lane & 15;
  const int koff = (lane >> 4) * 2;       // lanes 16-31 take K pair {2,3} of each step

  const float* aP  = &lds[(m0  + l15) * STRIDE + koff];
  const float* b0P = &lds[(n00 + l15) * STRIDE + koff];
  const float* b1P = &lds[(n01 + l15) * STRIDE + koff];

  v8f c0 = {};
  v8f c1 = {};
  #pragma unroll 4
  for (int k = 0; k < CK; k += 4) {
    v2f a  = *(const v2f*)(aP  + k);   // A 16x4: lane pair (K even, K odd)
    v2f b0 = *(const v2f*)(b0P + k);   // B 4x16: same striping (G symmetric: B rows = X rows)
    v2f b1 = *(const v2f*)(b1P + k);
    c0 = __builtin_amdgcn_wmma_f32_16x16x4_f32(false, a, false, b0, (short)0, c0, false, false);
    c1 = __builtin_amdgcn_wmma_f32_16x16x4_f32(false, a, false, b1, (short)0, c1, false, false);
  }

  // ---- per-lane partial: sum of all tile entries (constant indices only) ----
  float gs = 0.f;
  #pragma unroll
  for (int i = 0; i < 8; ++i) gs += c0[i];
  #pragma unroll
  for (int i = 0; i < 8; ++i) gs += c1[i];

  // ---- workgroup tree reduction (deterministic, no atomics) ----
  redA[tid] = gs;
  redB[tid] = tr;
  __syncthreads();
  #pragma unroll
  for (int s = THREADS / 2; s > 0; s >>= 1) {
    if (tid < s) {
      redA[tid] += redA[tid + s];
      redB[tid] += redB[tid + s];
    }
    __syncthreads();
  }
  if (tid == 0) {
    ws[2 * wg]     = redA[0];
    ws[2 * wg + 1] = redB[0];
  }
}

__global__ __launch_bounds__(THREADS)
void finalize_kernel(const float* __restrict__ ws, float* __restrict__ out) {
  __shared__ float redA[THREADS];
  __shared__ float redB[THREADS];
  const int tid = threadIdx.x;
  float gs = 0.f, tr = 0.f;
  for (int i = tid; i < NWG; i += THREADS) {
    gs += ws[2 * i];
    tr += ws[2 * i + 1];
  }
  redA[tid] = gs;
  redB[tid] = tr;
  __syncthreads();
  #pragma unroll
  for (int s = THREADS / 2; s > 0; s >>= 1) {
    if (tid < s) {
      redA[tid] += redA[tid + s];
      redB[tid] += redB[tid + s];
    }
    __syncthreads();
  }
  if (tid == 0) {
    const float Bf = (float)B_IMG;
    const float Df = (float)D_ELEM;
    // result = (sumG - B*trace) / (D * B * (B-1))
    out[0] = (redA[0] - Bf * redB[0]) / (Df * Bf * (Bf - 1.0f));
  }
}

extern "C" void kernel_launch(void* const* d_in, const int* in_sizes, int n_in,
                              void* d_out, int out_size, void* d_ws, size_t ws_size,
                              hipStream_t stream) {
  (void)in_sizes; (void)n_in; (void)out_size; (void)ws_size;
  const float* x   = (const float*)d_in[0];
  float*       ws  = (float*)d_ws;          // NWG * 2 floats, fully rewritten each call
  float*       out = (float*)d_out;

  gram_partial_kernel<<<dim3(NWG), dim3(THREADS), 0, stream>>>(x, ws);
  finalize_kernel<<<dim3(1), dim3(THREADS), 0, stream>>>(ws, out);
}